// YoloLoss_41712722379140
// MI455X (gfx1250) — compile-verified
//
#include <hip/hip_runtime.h>
#include <stdint.h>

#define SGRID 7
#define NBOX  5
#define NCLS  3
#define DROW  28                       // NBOX*5 + NCLS
#define BT    16384
#define NGT   8
#define CPI   (SGRID * SGRID)          // 49 cells per image
#define CELLS (BT * CPI)               // 802816
#define IMGSZ 448.0f
#define COORD_SCALE 5.0f
#define NOOBJ_SCALE 0.5f
#define TPB   256                      // cells per block in main kernel
#define NBLK  (CELLS / TPB)            // 3136 main-kernel blocks
#define RED_T 256                      // threads in final reduction kernel

__device__ __forceinline__ float sigmoidf_fast(float x) {
    return 1.0f / (1.0f + __expf(-x));
}

// ---------------------------------------------------------------------------
// Fused kernel: stream 256 rows (28 KB) into LDS via the CDNA5 async
// global->LDS datapath (non-temporal: data read exactly once), and while the
// DMA is in flight, rebuild the sparse targets for the <=7 batch images this
// block overlaps directly in LDS (one thread per image, boxes in order =>
// deterministic last-write-wins scatter, matching .at[idx].set()).
// Then per-cell loss, wave32 shuffle + LDS block reduction, one partial/block.
// ---------------------------------------------------------------------------
__global__ void __launch_bounds__(TPB)
yolo_loss_kernel(const float* __restrict__ output,
                 const float* __restrict__ bboxes,
                 const int*   __restrict__ labels,
                 float* __restrict__ partials) {
    __shared__ __align__(16) float smem[TPB * DROW];   // 7168 f32 = 28 KB
    __shared__ float4 ltb[TPB];                        // target box per local cell
    __shared__ int    llab[TPB];                       // target label per local cell
    __shared__ float  lflag[TPB];                      // object flag per local cell
    __shared__ float  red[(TPB / 32) * 3];             // per-wave partials

    const int cell0 = blockIdx.x * TPB;
    const float* gbase = output + (size_t)cell0 * DROW;
    const uint32_t lbase = (uint32_t)(uintptr_t)smem;  // LDS byte offset (low 32b of flat shared ptr)

    // 256 rows * 28 f32 = 7168 dwords = 1792 B128 transfers; 7 per thread.
#pragma unroll
    for (int j = 0; j < 7; ++j) {
        int idx4 = threadIdx.x + j * TPB;              // B128 index within tile
        unsigned long long gaddr =
            (unsigned long long)(uintptr_t)(gbase + (size_t)idx4 * 4);
        uint32_t laddr = lbase + (uint32_t)idx4 * 16u;
        asm volatile("global_load_async_to_lds_b128 %0, %1, off th:TH_LOAD_NT"
                     :: "v"(laddr), "v"(gaddr) : "memory");
    }

    // ---- build sparse targets in LDS while the async copy is in flight ----
    lflag[threadIdx.x] = 0.0f;
    __syncthreads();

    const int b0 = cell0 / CPI;                        // first image overlapping block
    const int b1 = (cell0 + TPB - 1) / CPI;            // last image (inclusive), b1-b0 <= 6
    if ((int)threadIdx.x <= b1 - b0) {
        int b = b0 + threadIdx.x;
        for (int n = 0; n < NGT; ++n) {                // serial => last write wins
            const float* p = bboxes + ((size_t)b * NGT + n) * 4;
            float x1 = p[0] * (1.0f / IMGSZ);
            float y1 = p[1] * (1.0f / IMGSZ);
            float x2 = p[2] * (1.0f / IMGSZ);
            float y2 = p[3] * (1.0f / IMGSZ);
            float w  = x2 - x1;
            float h  = y2 - y1;
            float cx = (x2 + x1) * 0.5f;
            float cy = (y2 + y1) * 0.5f;
            float gcx = fminf(fmaxf(ceilf(cx * (float)SGRID) - 1.0f, 0.0f), (float)(SGRID - 1));
            float gcy = fminf(fmaxf(ceilf(cy * (float)SGRID) - 1.0f, 0.0f), (float)(SGRID - 1));
            float dx = (cx - gcx * (1.0f / (float)SGRID)) * (float)SGRID;
            float dy = (cy - gcy * (1.0f / (float)SGRID)) * (float)SGRID;
            int idx = b * CPI + (int)gcy * SGRID + (int)gcx;   // global cell
            int li  = idx - cell0;                             // local cell
            if (li >= 0 && li < TPB) {                         // distinct images -> distinct cells
                lflag[li] = 1.0f;
                ltb[li]   = make_float4(dx, dy, w, h);
                llab[li]  = labels[b * NGT + n];
            }
        }
    }

    asm volatile("s_wait_asynccnt 0" ::: "memory");
    __syncthreads();

    // ---------------------------- per-cell loss ----------------------------
    const float* row = &smem[threadIdx.x * DROW];

    float sg[DROW];
#pragma unroll
    for (int i = 0; i < DROW; ++i) sg[i] = sigmoidf_fast(row[i]);

    float cl = 0.0f;   // coord loss
    float cf = 0.0f;   // conf  loss
    float cs = 0.0f;   // class loss

    if (lflag[threadIdx.x] != 1.0f) {
        // no-object cell: NOOBJ * sum_b (sigmoid(conf_b) - 0)^2
#pragma unroll
        for (int i = 0; i < NBOX; ++i) {
            float p = sg[5 * i + 4];
            cf += NOOBJ_SCALE * p * p;
        }
    } else {
        float4 tb = ltb[threadIdx.x];
        int lab = llab[threadIdx.x];

        float tx1 = tb.x - tb.z * 0.5f, ty1 = tb.y - tb.w * 0.5f;
        float tx2 = tb.x + tb.z * 0.5f, ty2 = tb.y + tb.w * 0.5f;
        float tarea = (tx2 - tx1) * (ty2 - ty1);

        float bx[NBOX], by[NBOX], bw[NBOX], bh[NBOX], bc[NBOX];
        int   best = 0;
        float bestiou = -1.0f;
#pragma unroll
        for (int i = 0; i < NBOX; ++i) {
            bx[i] = sg[5 * i + 0];
            by[i] = sg[5 * i + 1];
            bw[i] = sg[5 * i + 2];
            bh[i] = sg[5 * i + 3];
            bc[i] = sg[5 * i + 4];
            float hw = bw[i] * bw[i] * 0.5f;
            float hh = bh[i] * bh[i] * 0.5f;
            float px1 = bx[i] - hw, px2 = bx[i] + hw;
            float py1 = by[i] - hh, py2 = by[i] + hh;
            float iw = fmaxf(fminf(tx2, px2) - fmaxf(tx1, px1), 0.0f);
            float ih = fmaxf(fminf(ty2, py2) - fmaxf(ty1, py1), 0.0f);
            float inter = iw * ih;
            float parea = (px2 - px1) * (py2 - py1);
            float iou = inter / (tarea + parea - inter + 1e-8f);
            if (iou > bestiou) { bestiou = iou; best = i; }  // first-max, like jnp.argmax
        }
        float d0 = tb.x - bx[best];
        float d1 = tb.y - by[best];
        float d2 = sqrtf(tb.z) - bw[best];
        float d3 = sqrtf(tb.w) - bh[best];
        cl = COORD_SCALE * (d0 * d0 + d1 * d1 + d2 * d2 + d3 * d3);

        float e = 1.0f - bc[best];
        cf += e * e;

#pragma unroll
        for (int c = 0; c < NCLS; ++c) {
            float t = (c == lab) ? 1.0f : 0.0f;
            float d = sg[5 * NBOX + c] - t;
            cs += d * d;
        }
    }

    // wave32 shuffle reduction
#pragma unroll
    for (int off = 16; off > 0; off >>= 1) {
        cl += __shfl_down(cl, off, 32);
        cf += __shfl_down(cf, off, 32);
        cs += __shfl_down(cs, off, 32);
    }
    const int wv = threadIdx.x >> 5;
    if ((threadIdx.x & 31) == 0) {
        red[wv * 3 + 0] = cl;
        red[wv * 3 + 1] = cf;
        red[wv * 3 + 2] = cs;
    }
    __syncthreads();
    if (threadIdx.x == 0) {
        float a = 0.0f, b = 0.0f, c = 0.0f;
#pragma unroll
        for (int w = 0; w < TPB / 32; ++w) {
            a += red[w * 3 + 0];
            b += red[w * 3 + 1];
            c += red[w * 3 + 2];
        }
        partials[0 * NBLK + blockIdx.x] = a;   // coord
        partials[1 * NBLK + blockIdx.x] = b;   // conf
        partials[2 * NBLK + blockIdx.x] = c;   // class
    }
}

// ---------------------------------------------------------------------------
// Deterministic tree reduction of per-block partials -> 4 scalars.
// ---------------------------------------------------------------------------
__global__ void __launch_bounds__(RED_T)
yolo_reduce_kernel(const float* __restrict__ partials, float* __restrict__ out) {
    __shared__ float s0[RED_T], s1[RED_T], s2[RED_T];
    float a = 0.0f, b = 0.0f, c = 0.0f;
    for (int j = threadIdx.x; j < NBLK; j += RED_T) {
        a += partials[0 * NBLK + j];
        b += partials[1 * NBLK + j];
        c += partials[2 * NBLK + j];
    }
    s0[threadIdx.x] = a;
    s1[threadIdx.x] = b;
    s2[threadIdx.x] = c;
    __syncthreads();
    for (int off = RED_T / 2; off > 0; off >>= 1) {
        if (threadIdx.x < off) {
            s0[threadIdx.x] += s0[threadIdx.x + off];
            s1[threadIdx.x] += s1[threadIdx.x + off];
            s2[threadIdx.x] += s2[threadIdx.x + off];
        }
        __syncthreads();
    }
    if (threadIdx.x == 0) {
        float cl = s0[0], cf = s1[0], cs = s2[0];
        const float inv = 1.0f / (float)BT;
        out[0] = (cl + cf + cs) * inv;
        out[1] = cl * inv;
        out[2] = cf * inv;
        out[3] = cs * inv;
    }
}

// ---------------------------------------------------------------------------
extern "C" void kernel_launch(void* const* d_in, const int* in_sizes, int n_in,
                              void* d_out, int out_size, void* d_ws, size_t ws_size,
                              hipStream_t stream) {
    (void)in_sizes; (void)n_in; (void)out_size; (void)ws_size;

    const float* output    = (const float*)d_in[0];  // (bt, 7*7*28) f32
    const float* bboxes_gt = (const float*)d_in[1];  // (bt, 8, 4)   f32
    const int*   labels    = (const int*)  d_in[2];  // (bt, 8)      i32
    float*       out       = (float*)d_out;          // 4 scalars

    float* partials = (float*)d_ws;                  // 3 * NBLK f32, fully written each call

    yolo_loss_kernel<<<NBLK, TPB, 0, stream>>>(output, bboxes_gt, labels, partials);
    yolo_reduce_kernel<<<1, RED_T, 0, stream>>>(partials, out);
}